// dwt_calculation_37572373906117
// MI455X (gfx1250) — compile-verified
//
#include <hip/hip_runtime.h>

#define INV_SQRT2F 0.70710678118654752440f

// Native clang vector type: required by __builtin_nontemporal_* and
// guarantees global_load_b128 / global_store_b128 lowering.
typedef float v4f __attribute__((ext_vector_type(4)));

// Haar DWT along axis 1 of (8, 4096, 1024) fp32.
//   cA[b,r,c] = (x[b,2r,c] + x[b,2r+1,c]) * 1/sqrt(2)
//   cD[b,r,c] = (x[b,2r,c] - x[b,2r+1,c]) * 1/sqrt(2)
// Pure streaming kernel: 256 MiB of once-touched traffic, AI ~= 0.25 flop/B.
// -> B128 loads/stores, non-temporal hints, gfx1250 global_prefetch_b8 on the
//    next grid-stride iteration. No WMMA: there is no matmul here and the
//    VMEM path is the sole bottleneck (HBM floor ~11.5 us at 23.3 TB/s).
__global__ __launch_bounds__(256) void haar_dwt_f32_kernel(
    const float* __restrict__ x,
    float* __restrict__ cA,
    float* __restrict__ cD,
    int n4)   // total v4f elements per output plane (= N_in / 8)
{
    const v4f* __restrict__ x4 = reinterpret_cast<const v4f*>(x);
    v4f* __restrict__ a4 = reinterpret_cast<v4f*>(cA);
    v4f* __restrict__ d4 = reinterpret_cast<v4f*>(cD);

    const int stride = (int)(gridDim.x * blockDim.x);

    for (int g = (int)(blockIdx.x * blockDim.x + threadIdx.x); g < n4; g += stride) {
        // 1024 fp32 columns = 256 v4f per output row.
        const int row = g >> 8;          // output row index (b*2048 + r)
        const int c4  = g & 255;         // v4f column group
        const int e_idx = row * 512 + c4;   // even input row, v4f units (row*2048 floats)
        const int o_idx = e_idx + 256;      // odd input row (+1024 floats)

        // Speculative prefetch of the next grid-stride iteration's lines
        // (gfx1250 global_prefetch_b8; dropped silently if out of range).
        const int gn = g + stride;
        if (gn < n4) {
            const int rown = gn >> 8;
            const int c4n  = gn & 255;
            const v4f* pe = x4 + (rown * 512 + c4n);
            __builtin_prefetch((const void*)pe, 0, 3);
            __builtin_prefetch((const void*)(pe + 256), 0, 3);
        }

        // Once-touched data: non-temporal B128 loads.
        const v4f e = __builtin_nontemporal_load(x4 + e_idx);
        const v4f o = __builtin_nontemporal_load(x4 + o_idx);

        const v4f a = (e + o) * INV_SQRT2F;
        const v4f d = (e - o) * INV_SQRT2F;

        // Once-written results: non-temporal B128 stores.
        __builtin_nontemporal_store(a, a4 + g);
        __builtin_nontemporal_store(d, d4 + g);
    }
}

extern "C" void kernel_launch(void* const* d_in, const int* in_sizes, int n_in,
                              void* d_out, int out_size, void* d_ws, size_t ws_size,
                              hipStream_t stream) {
    (void)n_in; (void)out_size; (void)d_ws; (void)ws_size;

    const float* x = (const float*)d_in[0];
    const int N    = in_sizes[0];      // 8*4096*1024 = 33,554,432 fp32
    const int half = N / 2;            // elements per output plane
    const int n4   = half / 4;         // v4f elements per output plane

    float* cA = (float*)d_out;         // outputs concatenated: cA then cD
    float* cD = cA + half;

    const int block = 256;             // 8 wave32 per workgroup
    int blocks = (n4 + block - 1) / block;
    const int maxBlocks = 4096;        // grid-stride: ~4 iterations/thread,
                                       // gives the prefetch a real target
    if (blocks > maxBlocks) blocks = maxBlocks;
    if (blocks < 1) blocks = 1;

    haar_dwt_f32_kernel<<<blocks, block, 0, stream>>>(x, cA, cD, n4);
}